// AttenFFT_6794638262864
// MI455X (gfx1250) — compile-verified
//
#include <hip/hip_runtime.h>
#include <math.h>
#include <stdint.h>

#define BB 4
#define CC 64
#define NTOT 65536           // H*W
#define NHEADS 2
#define CHD 32
#define KSPLIT 64            // K-chunks for the Gram matmul (1024 each)
#define PLANE ((size_t)BB * CC * NTOT)   // 16,777,216 floats = 64 MB

typedef __attribute__((ext_vector_type(2))) float v2f;
typedef __attribute__((ext_vector_type(8))) float v8f;

// ---------------------------------------------------------------------------
// CDNA5 WMMA: D(16x16,f32) = A(16x4,f32) * B(4x16,f32) + C
// A frag: lane L<16 -> {A[M=L][K=0],A[M=L][K=1]}, lane L+16 -> K=2,3
// B frag: lane L<16 -> {B[K=0][N=L],B[K=1][N=L]}, lane L+16 -> K=2,3
// D: VGPR j, lanes 0-15 -> (M=j, N=lane); lanes 16-31 -> (M=j+8, N=lane-16)
// ---------------------------------------------------------------------------
__device__ __forceinline__ v8f wmma4(v2f a, v2f b, v8f c) {
  return __builtin_amdgcn_wmma_f32_16x16x4_f32(
      false, a, false, b, (short)0, c, false, false);
}

// ---------------------------------------------------------------------------
// Depthwise 3x3 conv (pad 1) producing q,k,v real planes in one x pass.
// ---------------------------------------------------------------------------
__global__ void dwconv3_kernel(const float* __restrict__ x,
                               const float* __restrict__ wq,
                               const float* __restrict__ wk,
                               const float* __restrict__ wv,
                               float* __restrict__ oq,
                               float* __restrict__ ok,
                               float* __restrict__ ov) {
  int pid = blockIdx.x * blockDim.x + threadIdx.x;
  int w = pid & 255;
  int h = (pid >> 8) & 255;
  int bc = pid >> 16;        // b*64 + c
  int c = bc & 63;
  float aq = 0.f, ak = 0.f, av = 0.f;
  for (int dy = 0; dy < 3; ++dy) {
    int yy = h + dy - 1;
    if (yy < 0 || yy > 255) continue;
    for (int dx = 0; dx < 3; ++dx) {
      int xx = w + dx - 1;
      if (xx < 0 || xx > 255) continue;
      float xv = x[((size_t)bc << 16) + yy * 256 + xx];
      int t = c * 9 + dy * 3 + dx;
      aq += xv * wq[t]; ak += xv * wk[t]; av += xv * wv[t];
    }
  }
  size_t o = ((size_t)bc << 16) + (h << 8) + w;
  oq[o] = aq; ok[o] = ak; ov[o] = av;
}

// ---------------------------------------------------------------------------
// 256-point complex FFT lines, Stockham autosort in LDS, 128 thr / line.
// line = blockIdx.x; img = line>>8; sub = line&255.
// base = img*65536 + sub*subStride ; element e at base + e*elemStride.
// sign: -1 fwd, +1 inv.  twidN>0: multiply out[e] by exp(sign*2pi*sub*e/twidN)
// (four-step twiddle for the 65536-point transform).  srcI==null -> imag 0.
// Loads are staged into LDS via the CDNA5 async gather path
// (GLOBAL_LOAD_ASYNC_TO_LDS_B32, ASYNCcnt) -- ideal for the stride-256
// column passes where consecutive elements are 1 KB apart.
// ---------------------------------------------------------------------------
__global__ void fft256_kernel(const float* __restrict__ srcR,
                              const float* __restrict__ srcI,
                              float* __restrict__ dstR,
                              float* __restrict__ dstI,
                              int subStride, int elemStride,
                              float sign, float scale, int twidN) {
  __shared__ float bre[2][256];
  __shared__ float bim[2][256];
  int line = blockIdx.x;
  int img = line >> 8;
  int sub = line & 255;
  size_t base = ((size_t)img << 16) + (size_t)sub * subStride;
  int t = threadIdx.x;   // 0..127
  for (int e = t; e < 256; e += 128) {
    size_t idx = base + (size_t)e * elemStride;
#if defined(__AMDGCN__)
    unsigned ldsR = (unsigned)(uintptr_t)(&bre[0][e]);   // addr[31:0] == LDS offset
    unsigned long long gaR = (unsigned long long)(uintptr_t)(srcR + idx);
    asm volatile("global_load_async_to_lds_b32 %0, %1, off"
                 :: "v"(ldsR), "v"(gaR) : "memory");
    if (srcI) {
      unsigned ldsI = (unsigned)(uintptr_t)(&bim[0][e]);
      unsigned long long gaI = (unsigned long long)(uintptr_t)(srcI + idx);
      asm volatile("global_load_async_to_lds_b32 %0, %1, off"
                   :: "v"(ldsI), "v"(gaI) : "memory");
    } else {
      bim[0][e] = 0.0f;
    }
#else
    bre[0][e] = srcR[idx];
    bim[0][e] = srcI ? srcI[idx] : 0.0f;
#endif
  }
#if defined(__AMDGCN__)
  asm volatile("s_wait_asynccnt 0" ::: "memory");
#endif
  __syncthreads();
  int cur = 0, n = 256, s = 1, ls = 0;
  #pragma unroll 1
  for (int st = 0; st < 8; ++st) {
    int m = n >> 1;
    int p = t >> ls;
    int q = t & (s - 1);
    int ia = q + s * p;
    int ib = ia + s * m;
    float ar = bre[cur][ia], ai = bim[cur][ia];
    float br = bre[cur][ib], bi = bim[cur][ib];
    float ang = sign * 6.28318530718f * (float)p / (float)n;
    float cs, sn; __sincosf(ang, &sn, &cs);
    int oa = q + s * (2 * p);
    int ob = oa + s;
    int nx = cur ^ 1;
    bre[nx][oa] = ar + br; bim[nx][oa] = ai + bi;
    float dr = ar - br, di = ai - bi;
    bre[nx][ob] = dr * cs - di * sn;
    bim[nx][ob] = dr * sn + di * cs;
    __syncthreads();
    cur = nx; s <<= 1; ++ls; n >>= 1;
  }
  for (int e = t; e < 256; e += 128) {
    float re = bre[cur][e], im = bim[cur][e];
    if (twidN > 0) {
      float ang = sign * 6.28318530718f * (float)(sub * e) / (float)twidN;
      float cs, sn; __sincosf(ang, &sn, &cs);
      float r2 = re * cs - im * sn;
      im = re * sn + im * cs; re = r2;
    }
    size_t idx = base + (size_t)e * elemStride;
    dstR[idx] = re * scale;
    dstI[idx] = im * scale;
  }
}

// ---------------------------------------------------------------------------
// Row L2 norms of q,k complex fields -> reciprocal scales.
// blockIdx: [0,512): field = blk>>8 (0=q,1=k), row = blk&255 (b*64+c).
// ---------------------------------------------------------------------------
__global__ void norm_kernel(const float* __restrict__ qr, const float* __restrict__ qi,
                            const float* __restrict__ kr, const float* __restrict__ ki,
                            float* __restrict__ qs, float* __restrict__ ks) {
  __shared__ float red[256];
  int blk = blockIdx.x;
  int field = blk >> 8;
  int row = blk & 255;
  const float* pr = field ? kr : qr;
  const float* pi = field ? ki : qi;
  size_t basep = (size_t)row * NTOT;
  float s = 0.f;
  for (int n = threadIdx.x; n < NTOT; n += 256) {
    float a = pr[basep + n], b = pi[basep + n];
    s += a * a + b * b;
  }
  red[threadIdx.x] = s;
  __syncthreads();
  for (int o = 128; o > 0; o >>= 1) {
    if ((int)threadIdx.x < o) red[threadIdx.x] += red[threadIdx.x + o];
    __syncthreads();
  }
  if (threadIdx.x == 0) {
    float sc = 1.0f / fmaxf(sqrtf(red[0]), 1e-12f);
    (field ? ks : qs)[row] = sc;
  }
}

// ---------------------------------------------------------------------------
// K-split Gram matmul: partial[kc][bh][c][d] = sum_{n in chunk} q[c][n]*k[d][n]
// (complex, unscaled).  Grid = 8 (b,h) * 64 K-chunks; 4 waves per block each
// own one 16x16 (c,d) tile (real+imag via 4 WMMA accumulators).
// 2048 waves stream the 256 MB of q/k -> enough parallelism for 23.3 TB/s.
// ---------------------------------------------------------------------------
__global__ void attn_wmma_kernel(const float* __restrict__ qr, const float* __restrict__ qi,
                                 const float* __restrict__ kr, const float* __restrict__ ki,
                                 float* __restrict__ pR, float* __restrict__ pI) {
  int kc = blockIdx.x & (KSPLIT - 1);
  int bh = blockIdx.x >> 6;            // b*2 + head
  int wave = threadIdx.x >> 5;         // 0..3
  int lane = threadIdx.x & 31;
  int c0 = (wave >> 1) * 16;
  int d0 = (wave & 1) * 16;
  int l15 = lane & 15;
  int ksel = (lane >> 4) * 2;
  const float* qrp = qr + (size_t)(bh * CHD) * NTOT;
  const float* qip = qi + (size_t)(bh * CHD) * NTOT;
  const float* krp = kr + (size_t)(bh * CHD) * NTOT;
  const float* kip = ki + (size_t)(bh * CHD) * NTOT;
  size_t rowA = (size_t)(c0 + l15) * NTOT + ksel;
  size_t rowB = (size_t)(d0 + l15) * NTOT + ksel;
  int kbeg = kc << 10;                 // 1024-wide chunk
  v8f srr = {}, sii = {}, sri = {}, sir = {};
  for (int k0 = kbeg; k0 < kbeg + 1024; k0 += 4) {
    v2f aR = *(const v2f*)(qrp + rowA + k0);
    v2f aI = *(const v2f*)(qip + rowA + k0);
    v2f bR = *(const v2f*)(krp + rowB + k0);
    v2f bI = *(const v2f*)(kip + rowB + k0);
    srr = wmma4(aR, bR, srr);
    sii = wmma4(aI, bI, sii);
    sri = wmma4(aR, bI, sri);
    sir = wmma4(aI, bR, sir);
    __builtin_prefetch(qrp + rowA + k0 + 512, 0, 1);   // global_prefetch_b8
    __builtin_prefetch(krp + rowB + k0 + 512, 0, 1);
  }
  #pragma unroll
  for (int j = 0; j < 8; ++j) {
    int c = c0 + j + ((lane >> 4) << 3);
    int d = d0 + l15;
    size_t o = ((size_t)(kc * 256 + bh * CHD + c)) * CHD + d;
    pR[o] = srr[j] - sii[j];
    pI[o] = sri[j] + sir[j];
  }
}

// ---------------------------------------------------------------------------
// Deterministic ordered reduction over the 64 K-chunks, then scale by
// 1/||q_c|| * 1/||k_d|| * temperature[h], then complex softmax over d.
// One wave per (bh,c) row.
// ---------------------------------------------------------------------------
__global__ void reduce_softmax_kernel(const float* __restrict__ pR,
                                      const float* __restrict__ pI,
                                      const float* __restrict__ qs,
                                      const float* __restrict__ ks,
                                      const float* __restrict__ temp,
                                      float* __restrict__ attnR,
                                      float* __restrict__ attnI) {
  int row = blockIdx.x;        // 0..255 : bh*32 + c
  int d = threadIdx.x;         // 0..31
  int bh = row >> 5;
  float sc = qs[row] * ks[bh * CHD + d] * temp[bh & 1];
  float aR = 0.f, aI = 0.f;
  for (int kc = 0; kc < KSPLIT; ++kc) {
    size_t o = ((size_t)(kc * 256 + row)) * CHD + d;
    aR += pR[o]; aI += pI[o];
  }
  aR *= sc; aI *= sc;
  // softmax (real part)
  float m = aR;
  for (int o = 16; o > 0; o >>= 1) m = fmaxf(m, __shfl_xor(m, o, 32));
  float e = __expf(aR - m);
  float s = e;
  for (int o = 16; o > 0; o >>= 1) s += __shfl_xor(s, o, 32);
  attnR[row * 32 + d] = e / s;
  // softmax (imag part)
  m = aI;
  for (int o = 16; o > 0; o >>= 1) m = fmaxf(m, __shfl_xor(m, o, 32));
  e = __expf(aI - m);
  s = e;
  for (int o = 16; o > 0; o >>= 1) s += __shfl_xor(s, o, 32);
  attnI[row * 32 + d] = e / s;
}

// ---------------------------------------------------------------------------
// out[c][n] = sum_d attn[c][d]*v[d][n] (complex). M=32,K=32,N=65536.
// 8 waves/block; each wave one 16x16 (c,n) tile; 8 K-steps * 4 complex WMMAs.
// ---------------------------------------------------------------------------
__global__ void outein_wmma_kernel(const float* __restrict__ attnR,
                                   const float* __restrict__ attnI,
                                   const float* __restrict__ vr,
                                   const float* __restrict__ vi,
                                   float* __restrict__ tr, float* __restrict__ ti) {
  int gw = blockIdx.x * 8 + (threadIdx.x >> 5);
  int lane = threadIdx.x & 31;
  int ntile = gw & 4095;
  int mi = (gw >> 12) & 1;
  int bh = gw >> 13;
  int c0 = mi * 16, n0 = ntile * 16;
  int l15 = lane & 15, ksel = (lane >> 4) * 2;
  const float* vrp = vr + (size_t)(bh * CHD) * NTOT;
  const float* vip = vi + (size_t)(bh * CHD) * NTOT;
  v8f srr = {}, sii = {}, sri = {}, sir = {};
  #pragma unroll
  for (int k0 = 0; k0 < 32; k0 += 4) {
    v2f aR = *(const v2f*)(attnR + (size_t)(bh * CHD + c0 + l15) * CHD + k0 + ksel);
    v2f aI = *(const v2f*)(attnI + (size_t)(bh * CHD + c0 + l15) * CHD + k0 + ksel);
    int r = k0 + ksel;
    v2f bR = { vrp[(size_t)r * NTOT + n0 + l15], vrp[(size_t)(r + 1) * NTOT + n0 + l15] };
    v2f bI = { vip[(size_t)r * NTOT + n0 + l15], vip[(size_t)(r + 1) * NTOT + n0 + l15] };
    srr = wmma4(aR, bR, srr);
    sii = wmma4(aI, bI, sii);
    sri = wmma4(aR, bI, sri);
    sir = wmma4(aI, bR, sir);
  }
  #pragma unroll
  for (int j = 0; j < 8; ++j) {
    int c = c0 + j + ((lane >> 4) << 3);
    size_t o = (size_t)(bh * CHD + c) * NTOT + n0 + l15;
    tr[o] = srr[j] - sii[j];
    ti[o] = sri[j] + sir[j];
  }
}

// ---------------------------------------------------------------------------
// Inverse FFT-32 along channel axis (in registers, Stockham, fully unrolled)
// + 1/32 scale + |.| + undo four-step digit permutation of the n axis:
// output n-index k = 256e+f reads source linear index j = 256f+e.
// ---------------------------------------------------------------------------
#define BSTAGE(SR, SI, DR, DI, NSZ, SST)                                      \
  { _Pragma("unroll") for (int p = 0; p < (NSZ) / 2; ++p) {                   \
      float ang = 6.28318530718f * (float)p / (float)(NSZ);                   \
      float csv, snv; __sincosf(ang, &snv, &csv);                             \
      _Pragma("unroll") for (int q = 0; q < (SST); ++q) {                     \
        float ar = SR[q + (SST) * p], ai = SI[q + (SST) * p];                 \
        float br = SR[q + (SST) * (p + (NSZ) / 2)];                           \
        float bi = SI[q + (SST) * (p + (NSZ) / 2)];                           \
        DR[q + (SST) * 2 * p] = ar + br;                                      \
        DI[q + (SST) * 2 * p] = ai + bi;                                      \
        float dr = ar - br, di = ai - bi;                                     \
        DR[q + (SST) * (2 * p + 1)] = dr * csv - di * snv;                    \
        DI[q + (SST) * (2 * p + 1)] = dr * snv + di * csv; } } }

__global__ void ifft32_abs_kernel(const float* __restrict__ tr,
                                  const float* __restrict__ ti,
                                  float* __restrict__ outA) {
  size_t pid = (size_t)blockIdx.x * 256 + threadIdx.x;  // B*H*N points
  int k = (int)(pid & 65535);
  int bh = (int)(pid >> 16);
  int e = k >> 8, f = k & 255;
  int j = (f << 8) + e;
  float ur[32], ui[32], wr[32], wi[32];
  #pragma unroll
  for (int ch = 0; ch < 32; ++ch) {
    size_t idx = (size_t)(bh * CHD + ch) * NTOT + j;
    ur[ch] = tr[idx]; ui[ch] = ti[idx];
  }
  BSTAGE(ur, ui, wr, wi, 32, 1)
  BSTAGE(wr, wi, ur, ui, 16, 2)
  BSTAGE(ur, ui, wr, wi, 8, 4)
  BSTAGE(wr, wi, ur, ui, 4, 8)
  BSTAGE(ur, ui, wr, wi, 2, 16)
  #pragma unroll
  for (int ch = 0; ch < 32; ++ch) {
    size_t o = (size_t)(bh * CHD + ch) * NTOT + k;
    outA[o] = sqrtf(wr[ch] * wr[ch] + wi[ch] * wi[ch]) * (1.0f / 32.0f);
  }
}

// ---------------------------------------------------------------------------
// 1x1 conv GEMM (M=64,K=64,N=65536 per batch) with fused epilogue:
// mode 0: +bias, BN, ReLU      mode 1: +bias, sigmoid
// ---------------------------------------------------------------------------
__global__ void conv1x1_wmma_kernel(const float* __restrict__ wmat,
                                    const float* __restrict__ bias,
                                    const float* __restrict__ src,
                                    float* __restrict__ dst,
                                    const float* __restrict__ bn_g,
                                    const float* __restrict__ bn_b,
                                    const float* __restrict__ bn_m,
                                    const float* __restrict__ bn_v,
                                    int mode) {
  int gw = blockIdx.x * 8 + (threadIdx.x >> 5);
  int lane = threadIdx.x & 31;
  int ntile = gw & 4095;
  int mi = (gw >> 12) & 3;
  int b = gw >> 14;
  int o0 = mi * 16, n0 = ntile * 16;
  int l15 = lane & 15, ksel = (lane >> 4) * 2;
  const float* srcb = src + (size_t)(b * 64) * NTOT;
  v8f acc = {};
  #pragma unroll 4
  for (int k0 = 0; k0 < 64; k0 += 4) {
    v2f aF = *(const v2f*)(wmat + (o0 + l15) * 64 + k0 + ksel);
    int r = k0 + ksel;
    v2f bF = { srcb[(size_t)r * NTOT + n0 + l15],
               srcb[(size_t)(r + 1) * NTOT + n0 + l15] };
    acc = wmma4(aF, bF, acc);
  }
  #pragma unroll
  for (int j = 0; j < 8; ++j) {
    int o = o0 + j + ((lane >> 4) << 3);
    float y = acc[j] + bias[o];
    if (mode == 0) {
      y = (y - bn_m[o]) * rsqrtf(bn_v[o] + 1e-5f) * bn_g[o] + bn_b[o];
      y = fmaxf(y, 0.0f);
    } else {
      y = 1.0f / (1.0f + __expf(-y));
    }
    dst[(size_t)(b * 64 + o) * NTOT + n0 + l15] = y;
  }
}

// ---------------------------------------------------------------------------
// Gate multiply and |.| elementwise
// ---------------------------------------------------------------------------
__global__ void gate_mul_kernel(float* __restrict__ xr, float* __restrict__ xi,
                                const float* __restrict__ g) {
  size_t i = (size_t)blockIdx.x * 256 + threadIdx.x;
  float gg = g[i];
  xr[i] *= gg; xi[i] *= gg;
}

__global__ void abs_kernel(const float* __restrict__ xr, const float* __restrict__ xi,
                           float* __restrict__ o) {
  size_t i = (size_t)blockIdx.x * 256 + threadIdx.x;
  o[i] = sqrtf(xr[i] * xr[i] + xi[i] * xi[i]);
}

// ---------------------------------------------------------------------------
// Final projection GEMM: M=64, K=128 (cat of branch A & B), N=65536 per batch.
// ---------------------------------------------------------------------------
__global__ void proj_wmma_kernel(const float* __restrict__ wproj,
                                 const float* __restrict__ srcA,
                                 const float* __restrict__ srcB,
                                 float* __restrict__ out) {
  int gw = blockIdx.x * 8 + (threadIdx.x >> 5);
  int lane = threadIdx.x & 31;
  int ntile = gw & 4095;
  int mi = (gw >> 12) & 3;
  int b = gw >> 14;
  int o0 = mi * 16, n0 = ntile * 16;
  int l15 = lane & 15, ksel = (lane >> 4) * 2;
  v8f acc = {};
  #pragma unroll 4
  for (int k0 = 0; k0 < 128; k0 += 4) {
    v2f aF = *(const v2f*)(wproj + (o0 + l15) * 128 + k0 + ksel);
    int r = k0 + ksel;
    const float* p0 = (r < 64) ? srcA + (size_t)(b * 64 + r) * NTOT
                               : srcB + (size_t)(b * 64 + r - 64) * NTOT;
    const float* p1 = (r + 1 < 64) ? srcA + (size_t)(b * 64 + r + 1) * NTOT
                                   : srcB + (size_t)(b * 64 + r + 1 - 64) * NTOT;
    v2f bF = { p0[n0 + l15], p1[n0 + l15] };
    acc = wmma4(aF, bF, acc);
  }
  #pragma unroll
  for (int j = 0; j < 8; ++j) {
    int o = o0 + j + ((lane >> 4) << 3);
    out[(size_t)(b * 64 + o) * NTOT + n0 + l15] = acc[j];
  }
}

// ---------------------------------------------------------------------------
extern "C" void kernel_launch(void* const* d_in, const int* in_sizes, int n_in,
                              void* d_out, int out_size, void* d_ws, size_t ws_size,
                              hipStream_t stream) {
  (void)in_sizes; (void)n_in; (void)out_size; (void)ws_size;
  const float* x     = (const float*)d_in[0];
  const float* w_q   = (const float*)d_in[1];
  const float* w_k   = (const float*)d_in[2];
  const float* w_v   = (const float*)d_in[3];
  const float* temp  = (const float*)d_in[4];
  const float* w1    = (const float*)d_in[5];
  const float* b1    = (const float*)d_in[6];
  const float* bn_g  = (const float*)d_in[7];
  const float* bn_b  = (const float*)d_in[8];
  const float* bn_m  = (const float*)d_in[9];
  const float* bn_v  = (const float*)d_in[10];
  const float* w2    = (const float*)d_in[11];
  const float* b2    = (const float*)d_in[12];
  const float* wproj = (const float*)d_in[13];
  float* out = (float*)d_out;
  float* ws = (float*)d_ws;

  float* qr  = ws + 0 * PLANE;  float* qi  = ws + 1 * PLANE;
  float* kr  = ws + 2 * PLANE;  float* ki  = ws + 3 * PLANE;
  float* vr  = ws + 4 * PLANE;  float* vi  = ws + 5 * PLANE;
  float* xfr = ws + 6 * PLANE;  float* xfi = ws + 7 * PLANE;
  float* arA = ws + 8 * PLANE;                 // |branch A| plane
  float* g1  = ws + 9 * PLANE;  float* g2  = ws + 10 * PLANE;
  float* bfB = ws + 11 * PLANE;                // |branch B| plane
  float* qsc = ws + 12 * PLANE;                // 256
  float* ksc = qsc + 256;                      // 256
  float* atR = ksc + 256;                      // 8192 (softmaxed attn, real)
  float* atI = atR + 8192;                     // 8192
  float* atPR = atI + 8192;                    // 64*8192 partials (real)
  float* atPI = atPR + (size_t)KSPLIT * 8192;  // 64*8192 partials (imag)

  const int LINES = BB * CC * 256;             // 65536 FFT lines per pass
  dim3 b256(256), b128(128), b32(32);

  // 1) depthwise convs -> real planes
  dwconv3_kernel<<<(int)(PLANE / 256), b256, 0, stream>>>(x, w_q, w_k, w_v, qr, kr, vr);

  // 2) forward fft2 over (h,w) for q,k,v (row pass over w, col pass over h)
  fft256_kernel<<<LINES, b128, 0, stream>>>(qr, nullptr, qr, qi, 256, 1, -1.f, 1.f, 0);
  fft256_kernel<<<LINES, b128, 0, stream>>>(qr, qi, qr, qi, 1, 256, -1.f, 1.f, 0);
  fft256_kernel<<<LINES, b128, 0, stream>>>(kr, nullptr, kr, ki, 256, 1, -1.f, 1.f, 0);
  fft256_kernel<<<LINES, b128, 0, stream>>>(kr, ki, kr, ki, 1, 256, -1.f, 1.f, 0);
  fft256_kernel<<<LINES, b128, 0, stream>>>(vr, nullptr, vr, vi, 256, 1, -1.f, 1.f, 0);
  fft256_kernel<<<LINES, b128, 0, stream>>>(vr, vi, vr, vi, 1, 256, -1.f, 1.f, 0);

  // 3) L2 norm reciprocal scales (folded into softmax pre-scale)
  norm_kernel<<<512, b256, 0, stream>>>(qr, qi, kr, ki, qsc, ksc);

  // 4) K-split Gram matmul via WMMA -> partials (2048 waves for bandwidth)
  attn_wmma_kernel<<<BB * NHEADS * KSPLIT, b128, 0, stream>>>(qr, qi, kr, ki, atPR, atPI);

  // 5) deterministic chunk reduction + scales + complex softmax
  reduce_softmax_kernel<<<BB * NHEADS * CHD, b32, 0, stream>>>(
      atPR, atPI, qsc, ksc, temp, atR, atI);

  // 6) out = attn @ v via WMMA (q planes reused as scratch)
  outein_wmma_kernel<<<8192, b256, 0, stream>>>(atR, atI, vr, vi, qr, qi);

  // 7) ifft over n=65536 (four-step: col FFT256 + twiddle, then row FFT256)
  fft256_kernel<<<LINES, b128, 0, stream>>>(qr, qi, qr, qi, 1, 256, +1.f, 1.f, 65536);
  fft256_kernel<<<LINES, b128, 0, stream>>>(qr, qi, qr, qi, 256, 1, +1.f, 1.f / 65536.f, 0);

  // 8) ifft over ch (32) + abs + digit-permutation fixup
  ifft32_abs_kernel<<<2048, b256, 0, stream>>>(qr, qi, arA);

  // 9) branch B: xf = fft2(x)
  fft256_kernel<<<LINES, b128, 0, stream>>>(x, nullptr, xfr, xfi, 256, 1, -1.f, 1.f, 0);
  fft256_kernel<<<LINES, b128, 0, stream>>>(xfr, xfi, xfr, xfi, 1, 256, -1.f, 1.f, 0);

  // 10-11) gated convs (WMMA GEMMs, fused BN/ReLU then sigmoid)
  conv1x1_wmma_kernel<<<8192, b256, 0, stream>>>(w1, b1, xfr, g1, bn_g, bn_b, bn_m, bn_v, 0);
  conv1x1_wmma_kernel<<<8192, b256, 0, stream>>>(w2, b2, g1, g2, nullptr, nullptr, nullptr, nullptr, 1);

  // 12) gate: xf *= g
  gate_mul_kernel<<<(int)(PLANE / 256), b256, 0, stream>>>(xfr, xfi, g2);

  // 13) ifft2 over (h,w), scale 1/65536
  fft256_kernel<<<LINES, b128, 0, stream>>>(xfr, xfi, xfr, xfi, 256, 1, +1.f, 1.f, 0);
  fft256_kernel<<<LINES, b128, 0, stream>>>(xfr, xfi, xfr, xfi, 1, 256, +1.f, 1.f / 65536.f, 0);

  // 14) abs
  abs_kernel<<<(int)(PLANE / 256), b256, 0, stream>>>(xfr, xfi, bfB);

  // 15) final projection GEMM over concatenated channels -> d_out
  proj_wmma_kernel<<<8192, b256, 0, stream>>>(wproj, arA, bfB, out);
}